// rotary_49881750176045
// MI455X (gfx1250) — compile-verified
//
#include <hip/hip_runtime.h>

typedef float v2f __attribute__((ext_vector_type(2)));
typedef float v8f __attribute__((ext_vector_type(8)));

#define SEQD  2048
#define HD    64
#define ROTN  32
#define HEADS 16
#define MSTR  66   // padded LDS row stride for transposed M (even -> 8B align;
                   // bank = (2*col + k) % 64 -> conflict-free b64 reads)

// ---------------------------------------------------------------------------
// Kernel 1: fold the Givens-rotation chain and r_matrix into one 64x64 matrix
//   M = T @ r_matrix, where  xf_rotated = xf @ T  (T built by applying the
//   column ops of the scan to the identity).
// One block of 64 threads; thread r owns row r of T (in LDS).
// ---------------------------------------------------------------------------
__global__ __launch_bounds__(64) void build_M_kernel(
    const float* __restrict__ thetas,
    const float* __restrict__ theta_scale,
    const float* __restrict__ r_matrix,
    const int*   __restrict__ pairs,
    float*       __restrict__ M)
{
    __shared__ float T[HD][HD];
    const int r = threadIdx.x;          // 0..63
    for (int c = 0; c < HD; ++c) T[r][c] = (c == r) ? 1.0f : 0.0f;

    const float ts = theta_scale[0];
    for (int k = 0; k < ROTN; ++k) {
        const int i = pairs[2 * k + 0];
        const int j = pairs[2 * k + 1];
        const float th = thetas[k] * ts;
        const float c = cosf(th), s = sinf(th);
        if (i == j) {
            // reference: col i set to ni, then overwritten with nj = xi*s
            T[r][i] = T[r][i] * s;
        } else {
            const float xi = T[r][i], xj = T[r][j];
            T[r][i] = xi * c + xj * s;
            T[r][j] = xj * c - xi * s;
        }
    }
    // M[r][c] = sum_k T[r][k] * R[k][c]
    for (int c = 0; c < HD; ++c) {
        float acc = 0.0f;
        for (int k = 0; k < HD; ++k) acc += T[r][k] * r_matrix[k * HD + c];
        M[r * HD + c] = acc;
    }
}

// ---------------------------------------------------------------------------
// Kernel 2: fused  out = RoPE( X @ M ) * sqrt(D)
// 256 threads = 8 wave32 per block; each wave handles one sequence position
// (16 heads -> a 16x64 row tile) via V_WMMA_F32_16X16X4_F32.
// M is staged TRANSPOSED in LDS so each lane's B pair (k, k+1 at fixed col)
// is one aligned ds_load_b64 straight into an even VGPR pair.
// ---------------------------------------------------------------------------
__global__ __launch_bounds__(256) void rope_gemm_kernel(
    const float* __restrict__ x,
    const float* __restrict__ inv_freq,
    const float* __restrict__ M,
    float*       __restrict__ out)
{
    __shared__ float MsT[HD * MSTR];    // ~16.5 KB: M transposed, padded stride
    __shared__ float Ys[8][16 * HD];    // 32 KB: per-wave D staging (16x64)
    __shared__ float SC[8][64];         // per-wave cos[0..31], sin[32..63]

    const int tid  = threadIdx.x;
    const int w    = tid >> 5;          // wave id 0..7
    const int lane = tid & 31;

    // cooperative load of M into LDS, transposed: MsT[c*MSTR + k] = M[k*HD + c]
    for (int t = tid; t < HD * HD; t += 256) {
        const int k = t >> 6;           // row of M
        const int c = t & 63;           // col of M
        MsT[c * MSTR + k] = M[t];
    }

    const int  p        = blockIdx.x * 8 + w;   // flat (b,s) position, 0..8191
    const int  s        = p & (SEQD - 1);       // sequence position
    const long row_base = (long)p * HEADS;      // first of 16 rows in xf

    // one sincos pair per lane covers all 16 heads of this position
    {
        const float ang = (float)s * inv_freq[lane];   // lane 0..31
        SC[w][lane]      = cosf(ang);
        SC[w][lane + 32] = sinf(ang);
    }

    // ---- A operands: 16 k-steps of the 16x4 f32 A layout ----
    // lanes 0-15 hold rows 0-15 with K = 4t,4t+1 ; lanes 16-31 hold K = 4t+2,4t+3
    const int arow  = lane & 15;
    const int khalf = (lane >> 4) << 1;                 // 0 or 2
    const float* xrow = x + (row_base + arow) * HD + khalf;
    v2f A[16];
#pragma unroll
    for (int t = 0; t < 16; ++t)
        A[t] = *(const v2f*)(xrow + 4 * t);             // global_load_b64

    __syncthreads();                                    // MsT + SC ready

    // ---- 64x V_WMMA_F32_16X16X4_F32: full 16x64 tile ----
    v8f acc[4];
#pragma unroll
    for (int n = 0; n < 4; ++n) {
        v8f c = {};
        const float* bcol = &MsT[(n * 16 + (lane & 15)) * MSTR + khalf];
#pragma unroll
        for (int t = 0; t < 16; ++t) {
            const v2f b = *(const v2f*)(bcol + 4 * t);  // ds_load_b64, no repack
            c = __builtin_amdgcn_wmma_f32_16x16x4_f32(
                    /*neg_a=*/false, A[t], /*neg_b=*/false, b,
                    /*c_mod=*/(short)0, c, /*reuse_a=*/false, /*reuse_b=*/false);
        }
        acc[n] = c;
    }

    // ---- stage D (16x16 f32 C/D layout) into this wave's LDS tile ----
    const int drow = (lane >> 4) << 3;                  // 0 or 8
    const int dcol = lane & 15;
#pragma unroll
    for (int n = 0; n < 4; ++n) {
#pragma unroll
        for (int v = 0; v < 8; ++v)
            Ys[w][(drow + v) * HD + n * 16 + dcol] = acc[n][v];
    }
    __syncthreads();

    // ---- RoPE: out[d]    = y[2d]*cos - y[2d+1]*sin   (d < 32)
    //           out[d+32] = y[2d]*sin + y[2d+1]*cos,  all * sqrt(1024)=32
    const float cs = SC[w][lane];
    const float sn = SC[w][lane + 32];
    float* obase = out + row_base * HD;
#pragma unroll
    for (int row = 0; row < 16; ++row) {
        const v2f y = *(const v2f*)&Ys[w][row * HD + 2 * lane];  // ds_load_b64
        const float oA = (y.x * cs - y.y * sn) * 32.0f;
        const float oB = (y.x * sn + y.y * cs) * 32.0f;
        obase[row * HD + lane]      = oA;
        obase[row * HD + lane + 32] = oB;
    }
}

// ---------------------------------------------------------------------------
extern "C" void kernel_launch(void* const* d_in, const int* in_sizes, int n_in,
                              void* d_out, int out_size, void* d_ws, size_t ws_size,
                              hipStream_t stream)
{
    const float* x           = (const float*)d_in[0];  // (4,2048,1024) f32
    const float* thetas      = (const float*)d_in[1];  // (32,) f32
    const float* theta_scale = (const float*)d_in[2];  // (1,) f32
    const float* r_matrix    = (const float*)d_in[3];  // (64,64) f32
    const float* inv_freq    = (const float*)d_in[4];  // (32,) f32
    const int*   pairs       = (const int*)d_in[5];    // (32,2) i32

    float* M = (float*)d_ws;                           // 64*64*4 = 16 KB scratch

    build_M_kernel<<<1, 64, 0, stream>>>(thetas, theta_scale, r_matrix, pairs, M);

    // 4*2048 = 8192 positions, 8 per block (one per wave32)
    rope_gemm_kernel<<<8192 / 8, 256, 0, stream>>>(x, inv_freq, M, (float*)d_out);
}